// shiftmlp_12867722018836
// MI455X (gfx1250) — compile-verified
//
#include <hip/hip_runtime.h>

// ---------------------------------------------------------------------------
// ShiftMLP block for MI455X (gfx1250, wave32, WMMA).
//   B=2, H=W=D=32, N=32768, Cin=160, Ch=640
//   out = fc2( shiftW( gelu( dwconv3( fc1( shiftH(x) ) ) ) ) )
// GEMMs use V_WMMA_F32_16X16X4_F32 (fp32 matrix path, matches fp32 reference).
// Shift handled branchlessly: clamped base pointer per chunk + 0/1 mask
// multiply, so EXEC stays all-1s and loads use immediate offsets.
// ---------------------------------------------------------------------------

typedef float v2f __attribute__((ext_vector_type(2)));
typedef float v8f __attribute__((ext_vector_type(8)));

#define HH   32
#define NTOK 32768            // H*W*D
#define MTOT 65536            // B * NTOK
#define CIN  160
#define CHID 640

__device__ __forceinline__ float gelu_exact(float v) {
    return 0.5f * v * (1.0f + erff(v * 0.70710678118654752f));
}

// ---------------------------------------------------------------------------
// Kernel 1: fused shift-H + fc1 GEMM  (M=65536, K=160, N=640)
// One wave computes a 16x64 output tile: 4 accumulators, A reused 4x.
// blockDim = 128 (4 waves), grid = (M/16/4, 640/64)
// ---------------------------------------------------------------------------
__global__ __launch_bounds__(128)
void fc1_shiftH_wmma(const float* __restrict__ x,
                     const float* __restrict__ w,     // (640,160)
                     const float* __restrict__ bias,  // (640)
                     float* __restrict__ out)         // (65536,640)
{
    const int lane   = threadIdx.x & 31;
    const int wave   = threadIdx.x >> 5;
    const int mtile  = blockIdx.x * 4 + wave;   // 0..4095
    const int ntBase = blockIdx.y * 64;         // 0..576

    const int row  = lane & 15;   // M-row for A, N-col for B/D
    const int half = lane >> 4;   // K-pair selector

    // Decode the token this lane's A-rows belong to.
    const int t  = mtile * 16 + row;
    const int bb = t >> 15;               // batch
    const int n  = t & (NTOK - 1);
    const int h  = n >> 10;
    const int wc = (n >> 5) & 31;
    const int d  = n & 31;

    // Per-chunk shifted A base pointers (clamped) + 0/1 validity masks.
    const float* apc[5];
    float amask[5];
    #pragma unroll
    for (int chunk = 0; chunk < 5; ++chunk) {
        const int  hsrc  = h + 2 - chunk;              // shift-H fold-in
        const bool valid = (unsigned)hsrc < 32u;
        const int  hcl   = valid ? hsrc : 0;
        apc[chunk] = x + (long)((bb << 15) | (hcl << 10) | (wc << 5) | d) * CIN
                       + half * 2;
        amask[chunk] = valid ? 1.0f : 0.0f;
    }

    // Per-subtile weight base pointers (lane K-offset folded in).
    const int o0 = ntBase + row;          // B col = lane&15
    const float* wp[4];
    #pragma unroll
    for (int j = 0; j < 4; ++j)
        wp[j] = w + (long)(o0 + j * 16) * CIN + half * 2;

    v8f acc[4] = {v8f{}, v8f{}, v8f{}, v8f{}};

    #pragma unroll
    for (int chunk = 0; chunk < 5; ++chunk) {
        const float* ap = apc[chunk];
        const float  m  = amask[chunk];
        #pragma unroll
        for (int q = 0; q < 8; ++q) {
            const int off = chunk * 32 + q * 4;        // constant after unroll
            v2f a = *(const v2f*)(ap + off);
            a.x *= m; a.y *= m;
            #pragma unroll
            for (int j = 0; j < 4; ++j) {
                const v2f bf = *(const v2f*)(wp[j] + off);
                acc[j] = __builtin_amdgcn_wmma_f32_16x16x4_f32(
                    false, a, false, bf, (short)0, acc[j], false, false);
            }
        }
    }

    // D layout: VGPR r -> row r (lanes 0-15) / row r+8 (lanes 16-31), col = lane&15
    #pragma unroll
    for (int r = 0; r < 8; ++r) {
        const int  mrow = r + half * 8;
        const long tout = (long)(mtile * 16 + mrow);
        #pragma unroll
        for (int j = 0; j < 4; ++j) {
            const int o = ntBase + j * 16 + row;
            out[tout * CHID + o] = acc[j][r] + bias[o];
        }
    }
}

// ---------------------------------------------------------------------------
// Kernel 2: depthwise 3x3x3 conv (SAME, zero pad) + bias + exact GELU
// One thread per (token, channel); consecutive threads -> consecutive channels.
// Interior voxels (82%) take a branch-free 27-tap FMA path.
// ---------------------------------------------------------------------------
__global__ __launch_bounds__(256)
void dwconv_gelu(const float* __restrict__ hin,   // (65536,640)
                 const float* __restrict__ w,     // (640,1,3,3,3)
                 const float* __restrict__ bias,  // (640)
                 float* __restrict__ out)         // (65536,640)
{
    const long idx = (long)blockIdx.x * 256 + threadIdx.x;
    if (idx >= (long)MTOT * CHID) return;

    const int  ch = (int)(idx % CHID);
    const long t  = idx / CHID;
    const int  bb = (int)(t >> 15);
    const int  n  = (int)(t & (NTOK - 1));
    const int  h  = n >> 10;
    const int  wc = (n >> 5) & 31;
    const int  d  = n & 31;

    float acc = bias[ch];
    const float* __restrict__ wk = w + (long)ch * 27;

    const bool interior = ((h - 1) | (30 - h) | (wc - 1) | (30 - wc) |
                           (d - 1) | (30 - d)) >= 0;

    if (interior) {
        const float* __restrict__ p = hin + (t - 1024 - 32 - 1) * CHID + ch;
        #pragma unroll
        for (int i = 0; i < 3; ++i)
            #pragma unroll
            for (int j = 0; j < 3; ++j)
                #pragma unroll
                for (int k = 0; k < 3; ++k)
                    acc = fmaf(wk[i * 9 + j * 3 + k],
                               p[(long)(i * 1024 + j * 32 + k) * CHID], acc);
    } else {
        #pragma unroll
        for (int i = 0; i < 3; ++i) {
            const int hh = h + i - 1;
            if ((unsigned)hh >= 32u) continue;
            #pragma unroll
            for (int j = 0; j < 3; ++j) {
                const int ww = wc + j - 1;
                if ((unsigned)ww >= 32u) continue;
                #pragma unroll
                for (int k = 0; k < 3; ++k) {
                    const int dd = d + k - 1;
                    if ((unsigned)dd >= 32u) continue;
                    const long tt = ((long)bb << 15) | (hh << 10) | (ww << 5) | dd;
                    acc = fmaf(wk[i * 9 + j * 3 + k], hin[tt * CHID + ch], acc);
                }
            }
        }
    }
    out[idx] = gelu_exact(acc);
}

// ---------------------------------------------------------------------------
// Kernel 3: fused shift-W + fc2 GEMM  (M=65536, K=640, N=160)
// One wave computes a 16x32 output tile: 2 accumulators.
// blockDim = 128 (4 waves), grid = (M/16/4, 160/32)
// ---------------------------------------------------------------------------
__global__ __launch_bounds__(128)
void fc2_shiftW_wmma(const float* __restrict__ g,     // (65536,640) gelu output
                     const float* __restrict__ w,     // (160,640)
                     const float* __restrict__ bias,  // (160)
                     float* __restrict__ out)         // (65536,160)
{
    const int lane   = threadIdx.x & 31;
    const int wave   = threadIdx.x >> 5;
    const int mtile  = blockIdx.x * 4 + wave;   // 0..4095
    const int ntBase = blockIdx.y * 32;         // 0..128

    const int row  = lane & 15;
    const int half = lane >> 4;

    const int t  = mtile * 16 + row;
    const int bb = t >> 15;
    const int n  = t & (NTOK - 1);
    const int h  = n >> 10;
    const int wc = (n >> 5) & 31;
    const int d  = n & 31;

    // Per-chunk shifted A base pointers (clamped) + 0/1 validity masks.
    const float* apc[5];
    float amask[5];
    #pragma unroll
    for (int chunk = 0; chunk < 5; ++chunk) {
        const int  wsrc  = wc + 2 - chunk;             // shift-W fold-in
        const bool valid = (unsigned)wsrc < 32u;
        const int  wcl   = valid ? wsrc : 0;
        apc[chunk] = g + (long)((bb << 15) | (h << 10) | (wcl << 5) | d) * CHID
                       + half * 2;
        amask[chunk] = valid ? 1.0f : 0.0f;
    }

    const int o0 = ntBase + row;
    const float* wp[2];
    #pragma unroll
    for (int j = 0; j < 2; ++j)
        wp[j] = w + (long)(o0 + j * 16) * CHID + half * 2;

    v8f acc[2] = {v8f{}, v8f{}};

    #pragma unroll
    for (int chunk = 0; chunk < 5; ++chunk) {
        const float* ap = apc[chunk];
        const float  m  = amask[chunk];
        #pragma unroll 8
        for (int q = 0; q < 32; ++q) {
            const int off = chunk * 128 + q * 4;
            v2f a = *(const v2f*)(ap + off);
            a.x *= m; a.y *= m;
            #pragma unroll
            for (int j = 0; j < 2; ++j) {
                const v2f bf = *(const v2f*)(wp[j] + off);
                acc[j] = __builtin_amdgcn_wmma_f32_16x16x4_f32(
                    false, a, false, bf, (short)0, acc[j], false, false);
            }
        }
    }

    #pragma unroll
    for (int r = 0; r < 8; ++r) {
        const int  mrow = r + half * 8;
        const long tout = (long)(mtile * 16 + mrow);
        #pragma unroll
        for (int j = 0; j < 2; ++j) {
            const int o = ntBase + j * 16 + row;
            out[tout * CIN + o] = acc[j][r] + bias[o];
        }
    }
}

// ---------------------------------------------------------------------------
// Host launcher
// ---------------------------------------------------------------------------
extern "C" void kernel_launch(void* const* d_in, const int* in_sizes, int n_in,
                              void* d_out, int out_size, void* d_ws, size_t ws_size,
                              hipStream_t stream) {
    const float* x     = (const float*)d_in[0];
    const float* fc1_w = (const float*)d_in[1];
    const float* fc1_b = (const float*)d_in[2];
    const float* dw_w  = (const float*)d_in[3];
    const float* dw_b  = (const float*)d_in[4];
    const float* fc2_w = (const float*)d_in[5];
    const float* fc2_b = (const float*)d_in[6];
    // d_in[7..9] = H, W, D (compile-time fixed to 32 here)

    float* out = (float*)d_out;
    float* ws0 = (float*)d_ws;                         // h  after fc1   (65536*640 f32)
    float* ws1 = ws0 + (size_t)MTOT * CHID;            // gelu(conv(h))  (65536*640 f32)

    // fc1 + shift-H: 4096 M-tiles / 4 waves per block, 10 N-blocks of 64
    fc1_shiftH_wmma<<<dim3(1024, 10), 128, 0, stream>>>(x, fc1_w, fc1_b, ws0);

    // depthwise conv + gelu over 65536*640 elements
    const long total = (long)MTOT * CHID;
    dwconv_gelu<<<dim3((unsigned)((total + 255) / 256)), 256, 0, stream>>>(ws0, dw_w, dw_b, ws1);

    // fc2 + shift-W: 4096 M-tiles / 4 waves per block, 5 N-blocks of 32
    fc2_shiftW_wmma<<<dim3(1024, 5), 128, 0, stream>>>(ws1, fc2_w, fc2_b, out);
}